// InterpBaselineEncoder_28613072126033
// MI455X (gfx1250) — compile-verified
//
#include <hip/hip_runtime.h>
#include <math.h>
#include <stdint.h>

// Problem constants (from reference): B,N,M,H,W = 8, 8192, 4096, 128, 128
#define BB 8
#define NN 8192
#define MM 4096
#define HH 128
#define WW 128

#define WAVES 4          // waves per block; each wave owns a 16-target tile
#define CHUNK 2048       // contexts staged in LDS per chunk (16 KB)

typedef __attribute__((ext_vector_type(2))) float v2f;
typedef __attribute__((ext_vector_type(8))) float v8f;

// ---------------------------------------------------------------------------
// Kernel 0: per-batch mean of yc_on_grid (fill value for OOB bilinear).
// ---------------------------------------------------------------------------
__global__ __launch_bounds__(256) void batch_mean_kernel(
    const float* __restrict__ yc_on, float* __restrict__ means) {
  __shared__ float red[256];
  const int b = blockIdx.x;
  const int tid = threadIdx.x;
  const float* p = yc_on + (size_t)b * (HH * WW);
  float s = 0.0f;
  for (int i = tid; i < HH * WW; i += 256) s += p[i];
  red[tid] = s;
  __syncthreads();
  for (int off = 128; off > 0; off >>= 1) {
    if (tid < off) red[tid] += red[tid + off];
    __syncthreads();
  }
  if (tid == 0) means[b] = red[0] * (1.0f / (float)(HH * WW));
}

// ---------------------------------------------------------------------------
// Kernel 1: WMMA-tiled nearest-neighbor + bilinear + sigmoid mix.
//   d^2(p,c) = [px, py, |p|^2, 1] . [-2cx, -2cy, 1, |c|^2]
//   -> one v_wmma_f32_16x16x4_f32 per 16x16 (target x context) tile.
//   Context staging uses the CDNA5 async copy engine
//   (global_load_async_to_lds_b64, tracked by ASYNCcnt).
// ---------------------------------------------------------------------------
__global__ __launch_bounds__(32 * WAVES) void nn_interp_kernel(
    const float* __restrict__ xc_off,   // (B,N,2)
    const float* __restrict__ yc_off,   // (B,N)
    const float* __restrict__ yc_on,    // (B,H,W)
    const float* __restrict__ xt,       // (B,M,2)
    const float* __restrict__ mix_logit,// scalar
    const float* __restrict__ means,    // (B) from kernel 0
    float* __restrict__ out)            // (B,M)
{
  __shared__ float2 s_ctx[CHUNK];
  __shared__ float  s_ytoff[WAVES * 16];

  const int b    = blockIdx.y;
  const int tid  = threadIdx.x;
  const int wave = tid >> 5;
  const int lane = tid & 31;
  const int col  = lane & 15;          // column (context) / row (target) in tile
  const bool hi  = lane >= 16;

  const int tileBase = (blockIdx.x * WAVES + wave) * 16;  // first target of this wave

  // --- A matrix: row m = [px, py, |p|^2, 1] for target tileBase+m ---
  const float2* xtp = (const float2*)(xt + (size_t)b * MM * 2);
  const float2 t = xtp[tileBase + col];
  const float tn2 = t.x * t.x + t.y * t.y;
  v2f A;
  A[0] = hi ? tn2  : t.x;   // lanes16-31: K=2 (|p|^2) ; lanes0-15: K=0 (px)
  A[1] = hi ? 1.0f : t.y;   // lanes16-31: K=3 (1)     ; lanes0-15: K=1 (py)

  float minv[8];
  int   mini[8];
#pragma unroll
  for (int r = 0; r < 8; ++r) { minv[r] = 3.0e38f; mini[r] = 0; }

  const float2* ctxp = (const float2*)(xc_off + (size_t)b * NN * 2);
  // AS(3) offset of the staging buffer = low 32 bits of the generic pointer.
  const uint32_t lds_base = (uint32_t)(uintptr_t)(&s_ctx[0]);

  for (int chunk = 0; chunk < NN; chunk += CHUNK) {
    __syncthreads();   // previous chunk fully consumed before overwrite

    // Async copy CHUNK float2 contexts global -> LDS (no VGPR round-trip).
    // Loop is divergence-free: CHUNK % (32*WAVES) == 0.
    for (int i = tid; i < CHUNK; i += 32 * WAVES) {
      const uint32_t lds_addr = lds_base + (uint32_t)i * 8u;
      const uint64_t gaddr    = (uint64_t)(uintptr_t)(ctxp + chunk + i);
      asm volatile("global_load_async_to_lds_b64 %0, %1, off"
                   :: "v"(lds_addr), "v"(gaddr)
                   : "memory");
    }
    asm volatile("s_wait_asynccnt 0" ::: "memory");  // own wave's copies done
    __syncthreads();                                 // all waves' copies visible

    for (int tpos = 0; tpos < CHUNK; tpos += 16) {
      // --- B matrix tile: col n = [-2cx, -2cy, 1, |c|^2] for context idx ---
      const float2 c = s_ctx[tpos + col];
      v2f Bv;
      Bv[0] = hi ? 1.0f                    : -2.0f * c.x;  // K=2 : K=0
      Bv[1] = hi ? (c.x * c.x + c.y * c.y) : -2.0f * c.y;  // K=3 : K=1

      v8f acc = {};
      acc = __builtin_amdgcn_wmma_f32_16x16x4_f32(
          /*neg_a=*/false, A, /*neg_b=*/false, Bv,
          /*c_mod=*/(short)0, acc, /*reuse_a=*/false, /*reuse_b=*/false);

      const int idx = chunk + tpos + col;   // context id held by this lane's column
#pragma unroll
      for (int r = 0; r < 8; ++r) {
        const bool lt = acc[r] < minv[r];
        minv[r] = lt ? acc[r] : minv[r];
        mini[r] = lt ? idx    : mini[r];
      }
    }
  }

  // --- argmin reduction across the 16 columns (16-lane halves) ---
#pragma unroll
  for (int r = 0; r < 8; ++r) {
#pragma unroll
    for (int m = 8; m >= 1; m >>= 1) {
      const float ov = __shfl_xor(minv[r], m, 32);
      const int   oi = __shfl_xor(mini[r], m, 32);
      const bool  lt = ov < minv[r];
      minv[r] = lt ? ov : minv[r];
      mini[r] = lt ? oi : mini[r];
    }
  }

  // Rows 0-7 live in lanes 0-15 (lane 0 representative), rows 8-15 in lanes 16-31.
  const float* yoffp = yc_off + (size_t)b * NN;
  if (lane == 0 || lane == 16) {
    const int rbase = hi ? 8 : 0;
#pragma unroll
    for (int r = 0; r < 8; ++r)
      s_ytoff[wave * 16 + rbase + r] = yoffp[mini[r]];
  }
  __syncthreads();

  // --- bilinear interp + mix: lanes 0-15 each finalize one target ---
  if (lane < 16) {
    const float ytoff = s_ytoff[wave * 16 + lane];

    const float fx = t.x * (float)(HH - 1);
    const float fy = t.y * (float)(WW - 1);
    int ix = (int)floorf(fx); ix = ix < 0 ? 0 : (ix > HH - 2 ? HH - 2 : ix);
    int iy = (int)floorf(fy); iy = iy < 0 ? 0 : (iy > WW - 2 ? WW - 2 : iy);
    const float wx = fx - (float)ix;
    const float wy = fy - (float)iy;

    const float* g = yc_on + (size_t)b * HH * WW;
    const float v00 = g[ix * WW + iy];
    const float v01 = g[ix * WW + iy + 1];
    const float v10 = g[(ix + 1) * WW + iy];
    const float v11 = g[(ix + 1) * WW + iy + 1];
    const float v = (1.0f - wx) * (1.0f - wy) * v00 + (1.0f - wx) * wy * v01 +
                    wx * (1.0f - wy) * v10 + wx * wy * v11;

    const bool oob = (t.x < 0.0f) || (t.x > 1.0f) || (t.y < 0.0f) || (t.y > 1.0f);
    const float yton = oob ? means[b] : v;

    const float mix = 1.0f / (1.0f + __expf(-mix_logit[0]));
    out[(size_t)b * MM + tileBase + lane] = mix * ytoff + (1.0f - mix) * yton;
  }
}

// ---------------------------------------------------------------------------
extern "C" void kernel_launch(void* const* d_in, const int* in_sizes, int n_in,
                              void* d_out, int out_size, void* d_ws, size_t ws_size,
                              hipStream_t stream) {
  (void)in_sizes; (void)n_in; (void)out_size; (void)ws_size;
  const float* xc_off = (const float*)d_in[0];  // (B,N,2)
  const float* yc_off = (const float*)d_in[1];  // (B,N)
  // d_in[2] = xc_on_grid: analytic linspace grid, recomputed on the fly
  const float* yc_on  = (const float*)d_in[3];  // (B,H,W)
  const float* xtp    = (const float*)d_in[4];  // (B,M,2)
  const float* mix    = (const float*)d_in[5];  // scalar
  float* out   = (float*)d_out;
  float* means = (float*)d_ws;                  // 8 floats of scratch

  batch_mean_kernel<<<dim3(BB), dim3(256), 0, stream>>>(yc_on, means);

  dim3 grid(MM / (16 * WAVES), BB);
  nn_interp_kernel<<<grid, dim3(32 * WAVES), 0, stream>>>(
      xc_off, yc_off, yc_on, xtp, mix, means, out);
}